// QueryGuidedScaleGate_89696097009919
// MI455X (gfx1250) — compile-verified
//
#include <hip/hip_runtime.h>
#include <hip/hip_bf16.h>

typedef float v2f __attribute__((ext_vector_type(2)));
typedef float v8f __attribute__((ext_vector_type(8)));

#define BATCH   8
#define CCH     256
#define HDIM    256
#define TOPK    300
#define HW0     15200
#define HW1     3800
#define HW2     950
#define NTOK    19950      /* HW0+HW1+HW2 */
#define QIN     515
#define QPAD    516
#define ROWS    (BATCH*TOPK)   /* 2400 */
#define TOT4    1276800        /* CCH*NTOK/4 per batch */
#define INVK    (1.0f/300.0f)

// ---------------------------------------------------------------- score pass
// s[b,t] = dot_c f_l[b,c,p] * conv_w[c] + conv_b   (coalesced over p)
__global__ void score_kernel(const float* __restrict__ f0, const float* __restrict__ f1,
                             const float* __restrict__ f2, const float* __restrict__ conv_w,
                             const float* __restrict__ conv_b, float* __restrict__ scores) {
  int t = blockIdx.x * blockDim.x + threadIdx.x;
  int b = blockIdx.y;
  if (t >= NTOK) return;
  int p, HW; const float* f;
  if (t < HW0)            { p = t;            HW = HW0; f = f0; }
  else if (t < HW0 + HW1) { p = t - HW0;      HW = HW1; f = f1; }
  else                    { p = t - HW0 - HW1; HW = HW2; f = f2; }
  const float* base = f + (size_t)b * CCH * HW + p;
  float s = 0.0f;
#pragma unroll 8
  for (int c = 0; c < CCH; ++c) s += base[(size_t)c * HW] * conv_w[c];
  scores[b * NTOK + t] = s + conv_b[0];
}

// ---------------------------------------------------------------- top-k (radix select)
__device__ __forceinline__ unsigned fkey(float v) {
  unsigned u = __float_as_uint(v);
  return (u & 0x80000000u) ? ~u : (u | 0x80000000u);   // monotone: bigger key = bigger float
}

__global__ void topk_kernel(const float* __restrict__ scores,
                            int* __restrict__ sel_idx, float* __restrict__ sel_val) {
  int b = blockIdx.x;
  const float* s = scores + b * NTOK;
  __shared__ int hist[256];
  __shared__ unsigned sh_prefix, sh_mask;
  __shared__ int sh_remK, sh_selCount, sh_eqTicket;
  int tid = threadIdx.x;   // 256 threads
  if (tid == 0) { sh_prefix = 0u; sh_mask = 0u; sh_remK = TOPK; sh_selCount = 0; sh_eqTicket = 0; }
  __syncthreads();
  for (int r = 3; r >= 0; --r) {
    hist[tid] = 0;
    __syncthreads();
    unsigned prefix = sh_prefix, mask = sh_mask;
    for (int i = tid; i < NTOK; i += 256) {
      unsigned u = fkey(s[i]);
      if ((u & mask) == prefix) atomicAdd(&hist[(u >> (8 * r)) & 0xFFu], 1);
    }
    __syncthreads();
    if (tid == 0) {
      int remK = sh_remK, bsel = 0;
      for (int bb = 255; bb >= 0; --bb) {
        if (hist[bb] >= remK) { bsel = bb; break; }
        remK -= hist[bb];
      }
      sh_remK = remK;
      sh_prefix = prefix | ((unsigned)bsel << (8 * r));
      sh_mask = mask | (0xFFu << (8 * r));
    }
    __syncthreads();
  }
  unsigned thr = sh_prefix;
  int needEq = sh_remK;
  for (int i = tid; i < NTOK; i += 256) {
    float v = s[i];
    unsigned u = fkey(v);
    bool take = false;
    if (u > thr) take = true;
    else if (u == thr) { if (atomicAdd(&sh_eqTicket, 1) < needEq) take = true; }
    if (take) {
      int slot = atomicAdd(&sh_selCount, 1);
      sel_idx[b * TOPK + slot] = i;
      sel_val[b * TOPK + slot] = v;
    }
  }
}

// ---------------------------------------------------------------- build q rows [2400, 516]
__global__ void gather_kernel(const float* __restrict__ f0, const float* __restrict__ f1,
                              const float* __restrict__ f2, const float* __restrict__ level_emb,
                              const int* __restrict__ sel_idx, const float* __restrict__ sel_val,
                              float* __restrict__ q) {
  int row = blockIdx.x;            // 0..2399
  int b = row / TOPK;
  int t = sel_idx[row];
  int lvl, p, HW, H, W; const float* f;
  if (t < HW0)            { lvl = 0; p = t;             HW = HW0; H = 100; W = 152; f = f0; }
  else if (t < HW0 + HW1) { lvl = 1; p = t - HW0;       HW = HW1; H = 50;  W = 76;  f = f1; }
  else                    { lvl = 2; p = t - HW0 - HW1; HW = HW2; H = 25;  W = 38;  f = f2; }
  float* qrow = q + (size_t)row * QPAD;
  for (int c = threadIdx.x; c < CCH; c += blockDim.x)
    qrow[c] = f[((size_t)(b * CCH + c)) * HW + p];
  for (int c = threadIdx.x; c < HDIM; c += blockDim.x)
    qrow[CCH + c] = level_emb[lvl * HDIM + c];
  if (threadIdx.x == 0) {
    int x = p % W, y = p / W;
    qrow[512] = -1.0f + 2.0f * (float)x / (float)(W - 1);
    qrow[513] = -1.0f + 2.0f * (float)y / (float)(H - 1);
    qrow[514] = 1.0f / (1.0f + expf(-sel_val[row]));
    qrow[515] = 0.0f;
  }
}

// ---------------------------------------------------------------- pad w1 / gate head, zero alpha
__global__ void prep_kernel(const float* __restrict__ w1, const float* __restrict__ gate_w,
                            const float* __restrict__ gate_b, float* __restrict__ w1p,
                            float* __restrict__ gwp, float* __restrict__ gbp,
                            float* __restrict__ alpha) {
  int i = blockIdx.x * blockDim.x + threadIdx.x;
  if (i < QPAD * HDIM) {
    w1p[i] = (i < QIN * HDIM) ? w1[i] : 0.0f;              // zero pad row 515
  } else if (i < QPAD * HDIM + HDIM * 16) {
    int j = i - QPAD * HDIM, k = j >> 4, n = j & 15;
    gwp[j] = (n < 8) ? gate_w[k * 8 + n] : 0.0f;           // [256,8] -> [256,16]
  } else if (i < QPAD * HDIM + HDIM * 16 + 16) {
    int n = i - QPAD * HDIM - HDIM * 16;
    gbp[n] = (n < 8) ? gate_b[n] : 0.0f;
  } else if (i < QPAD * HDIM + HDIM * 16 + 16 + BATCH * 3) {
    alpha[i - QPAD * HDIM - HDIM * 16 - 16] = 0.0f;
  }
}

// ---------------------------------------------------------------- f32 WMMA GEMM
// one wave per 16x16 output tile; A [M,Ksz] row-major (lda), B [Ksz,N] row-major (ldb)
__global__ void wmma_gemm(const float* __restrict__ A, int lda,
                          const float* __restrict__ Bm, int ldb,
                          const float* __restrict__ bias, float* __restrict__ Out, int ldo,
                          int Ksz, int relu) {
  int mt = blockIdx.x, nt = blockIdx.y;
  int lane = threadIdx.x;       // 32 threads, EXEC all ones
  int half = lane >> 4;         // 0: K+{0,1}  1: K+{2,3}
  int l15 = lane & 15;
  int koff = half * 2;
  int col = nt * 16 + l15;
  const float* arow = A + (size_t)(mt * 16 + l15) * lda;
  v8f acc = {};
  for (int k0 = 0; k0 < Ksz; k0 += 4) {
    v2f a, b;
    a.x = arow[k0 + koff];
    a.y = arow[k0 + koff + 1];
    b.x = Bm[(size_t)(k0 + koff) * ldb + col];
    b.y = Bm[(size_t)(k0 + koff + 1) * ldb + col];
    acc = __builtin_amdgcn_wmma_f32_16x16x4_f32(false, a, false, b, (short)0, acc, false, false);
  }
  float bv = bias[col];
#pragma unroll
  for (int j = 0; j < 8; ++j) {
    int row = mt * 16 + j + half * 8;
    float v = acc[j] + bv;
    if (relu) v = fmaxf(v, 0.0f);
    Out[(size_t)row * ldo + col] = v;
  }
}

// ---------------------------------------------------------------- softmax over 3 gate logits, mean over K
__global__ void gate_softmax_kernel(const float* __restrict__ logits, float* __restrict__ alpha) {
  int t = blockIdx.x * blockDim.x + threadIdx.x;
  if (t >= ROWS) return;
  int b = t / TOPK;
  float l0 = logits[t * 16 + 0], l1 = logits[t * 16 + 1], l2 = logits[t * 16 + 2];
  float m = fmaxf(l0, fmaxf(l1, l2));
  float e0 = expf(l0 - m), e1 = expf(l1 - m), e2 = expf(l2 - m);
  float inv = 1.0f / (e0 + e1 + e2);
  atomicAdd(&alpha[b * 3 + 0], e0 * inv * INVK);
  atomicAdd(&alpha[b * 3 + 1], e1 * inv * INVK);
  atomicAdd(&alpha[b * 3 + 2], e2 * inv * INVK);
}

// ---------------------------------------------------------------- scale + copy (float4, bandwidth pass)
__global__ void scale_kernel(const float* __restrict__ src, float* __restrict__ out,
                             const float* __restrict__ alpha, const float* __restrict__ rs,
                             int lvl, int hw, int off4) {
  int nvec_per_b = CCH * hw / 4;
  long long i = (long long)blockIdx.x * blockDim.x + threadIdx.x;
  if (i >= (long long)BATCH * nvec_per_b) return;
  int b = (int)(i / nvec_per_b);
  int inner = (int)(i % nvec_per_b);
  float scale = 1.0f + rs[0] * alpha[b * 3 + lvl];
  float4 v = ((const float4*)src)[i];
  v.x *= scale; v.y *= scale; v.z *= scale; v.w *= scale;
  ((float4*)out)[(long long)b * TOT4 + off4 + inner] = v;
}

// ----------------------------------------------------------------
static inline size_t alignup(size_t x) { return (x + 255) & ~(size_t)255; }

extern "C" void kernel_launch(void* const* d_in, const int* in_sizes, int n_in,
                              void* d_out, int out_size, void* d_ws, size_t ws_size,
                              hipStream_t stream) {
  const float* f0      = (const float*)d_in[0];
  const float* f1      = (const float*)d_in[1];
  const float* f2      = (const float*)d_in[2];
  const float* conv_w  = (const float*)d_in[3];
  const float* conv_b  = (const float*)d_in[4];
  const float* lvl_emb = (const float*)d_in[5];
  const float* w1      = (const float*)d_in[6];
  const float* b1      = (const float*)d_in[7];
  const float* w2      = (const float*)d_in[8];
  const float* b2      = (const float*)d_in[9];
  const float* gate_w  = (const float*)d_in[10];
  const float* gate_b  = (const float*)d_in[11];
  const float* rs      = (const float*)d_in[12];
  float* out = (float*)d_out;

  char* ws = (char*)d_ws;
  size_t off = 0;
  float* scores  = (float*)(ws + off); off = alignup(off + (size_t)BATCH * NTOK * 4);
  int*   sel_idx = (int*)  (ws + off); off = alignup(off + (size_t)ROWS * 4);
  float* sel_val = (float*)(ws + off); off = alignup(off + (size_t)ROWS * 4);
  float* q       = (float*)(ws + off); off = alignup(off + (size_t)ROWS * QPAD * 4);
  float* w1p     = (float*)(ws + off); off = alignup(off + (size_t)QPAD * HDIM * 4);
  float* gwp     = (float*)(ws + off); off = alignup(off + (size_t)HDIM * 16 * 4);
  float* gbp     = (float*)(ws + off); off = alignup(off + (size_t)16 * 4);
  float* h1      = (float*)(ws + off); off = alignup(off + (size_t)ROWS * HDIM * 4);
  float* h2      = (float*)(ws + off); off = alignup(off + (size_t)ROWS * HDIM * 4);
  float* logits  = (float*)(ws + off); off = alignup(off + (size_t)ROWS * 16 * 4);
  float* alpha   = (float*)(ws + off); off = alignup(off + (size_t)BATCH * 3 * 4);
  (void)ws_size; (void)in_sizes; (void)n_in; (void)out_size;

  // 1. objectness scores (163 MB streaming pass)
  score_kernel<<<dim3((NTOK + 255) / 256, BATCH), 256, 0, stream>>>(f0, f1, f2, conv_w, conv_b, scores);
  // 2. per-batch top-300 via 8-bit radix select (order-invariant downstream)
  topk_kernel<<<BATCH, 256, 0, stream>>>(scores, sel_idx, sel_val);
  // 3. pad weights / zero alpha
  {
    int total = QPAD * HDIM + HDIM * 16 + 16 + BATCH * 3;
    prep_kernel<<<(total + 255) / 256, 256, 0, stream>>>(w1, gate_w, gate_b, w1p, gwp, gbp, alpha);
  }
  // 4. build q [2400, 516]
  gather_kernel<<<ROWS, 128, 0, stream>>>(f0, f1, f2, lvl_emb, sel_idx, sel_val, q);
  // 5. MLP + gate head with f32 WMMA (16x16x4)
  wmma_gemm<<<dim3(ROWS / 16, HDIM / 16), 32, 0, stream>>>(q, QPAD, w1p, HDIM, b1, h1, HDIM, QPAD, 1);
  wmma_gemm<<<dim3(ROWS / 16, HDIM / 16), 32, 0, stream>>>(h1, HDIM, w2, HDIM, b2, h2, HDIM, HDIM, 1);
  wmma_gemm<<<dim3(ROWS / 16, 1), 32, 0, stream>>>(h2, HDIM, gwp, 16, gbp, logits, 16, HDIM, 0);
  // 6. softmax + mean -> alpha[b,3]
  gate_softmax_kernel<<<(ROWS + 255) / 256, 256, 0, stream>>>(logits, alpha);
  // 7. scale + copy out (float4; the dominant 326 MB pass)
  {
    long long v0 = (long long)BATCH * CCH * HW0 / 4;
    long long v1 = (long long)BATCH * CCH * HW1 / 4;
    long long v2 = (long long)BATCH * CCH * HW2 / 4;
    scale_kernel<<<(int)((v0 + 255) / 256), 256, 0, stream>>>(f0, out, alpha, rs, 0, HW0, 0);
    scale_kernel<<<(int)((v1 + 255) / 256), 256, 0, stream>>>(f1, out, alpha, rs, 1, HW1, CCH * HW0 / 4);
    scale_kernel<<<(int)((v2 + 255) / 256), 256, 0, stream>>>(f2, out, alpha, rs, 2, HW2, CCH * (HW0 + HW1) / 4);
  }
}